// ComplexConvolutionalMultiheadAttention_3633542332461
// MI455X (gfx1250) — compile-verified
//
#include <hip/hip_runtime.h>
#include <math.h>

// ---------------- problem constants ----------------
#define NH   8          // heads
#define CHN  128        // channels
#define IH   48
#define IW   48
#define NTOK (IH*IW)    // 2304 tokens
#define HD   16         // head dim
#define HSZ  (NH*NTOK*HD)   // 294912 = one head-major plane (== CHN*NTOK)

typedef __attribute__((ext_vector_type(2))) float v2f;
typedef __attribute__((ext_vector_type(8))) float v8f;

// D = A(16x4) * B(4x16) + C  in fp32  (CDNA5 V_WMMA_F32_16X16X4_F32)
__device__ __forceinline__ v8f wmma_k4(v2f a, v2f b, v8f c) {
    return __builtin_amdgcn_wmma_f32_16x16x4_f32(
        /*neg_a=*/false, a, /*neg_b=*/false, b,
        /*c_mod=*/(short)0, c, /*reuse_a=*/false, /*reuse_b=*/false);
}

// ============================================================
// Kernel 1: complex 3x3 conv (pad 1) for q,k,v.
// Outputs written in head-major layout: out[((o>>4)*NTOK + p)*HD + (o&15)]
// grid: (NTOK/256, CHN, 3)  block: 256
// ============================================================
__global__ void conv_qkv_kernel(
    const float* __restrict__ xr, const float* __restrict__ xi,
    const float* __restrict__ wqr, const float* __restrict__ wqi,
    const float* __restrict__ bqr, const float* __restrict__ bqi,
    const float* __restrict__ wkr, const float* __restrict__ wki,
    const float* __restrict__ bkr, const float* __restrict__ bki,
    const float* __restrict__ wvr, const float* __restrict__ wvi,
    const float* __restrict__ bvr, const float* __restrict__ bvi,
    float* __restrict__ qr, float* __restrict__ qi,
    float* __restrict__ kr, float* __restrict__ ki,
    float* __restrict__ vr, float* __restrict__ vi)
{
    const int sel = blockIdx.z;             // 0=q,1=k,2=v
    const int o   = blockIdx.y;             // output channel
    const int p   = blockIdx.x * blockDim.x + threadIdx.x;   // spatial
    const int y = p / IW, x = p % IW;

    const float* wr; const float* wi; const float* br; const float* bi;
    float* outr; float* outi;
    if (sel == 0)      { wr=wqr; wi=wqi; br=bqr; bi=bqi; outr=qr; outi=qi; }
    else if (sel == 1) { wr=wkr; wi=wki; br=bkr; bi=bki; outr=kr; outi=ki; }
    else               { wr=wvr; wi=wvi; br=bvr; bi=bvi; outr=vr; outi=vi; }

    float accr = br[o];
    float acci = bi[o];
    for (int i = 0; i < CHN; ++i) {
        const float* xrp = xr + i * (IH*IW);
        const float* xip = xi + i * (IH*IW);
        const float* wrp = wr + (o * CHN + i) * 9;
        const float* wip = wi + (o * CHN + i) * 9;
        #pragma unroll
        for (int kh = 0; kh < 3; ++kh) {
            const int yy = y + kh - 1;
            if (yy < 0 || yy >= IH) continue;
            #pragma unroll
            for (int kw = 0; kw < 3; ++kw) {
                const int xx = x + kw - 1;
                if (xx < 0 || xx >= IW) continue;
                const float xrv = xrp[yy*IW + xx];
                const float xiv = xip[yy*IW + xx];
                const float wrv = wrp[kh*3 + kw];
                const float wiv = wip[kh*3 + kw];
                accr = fmaf(xrv, wrv, accr);
                accr = fmaf(-xiv, wiv, accr);
                acci = fmaf(xrv, wiv, acci);
                acci = fmaf(xiv, wrv, acci);
            }
        }
    }
    const int idx = ((o >> 4) * NTOK + p) * HD + (o & 15);
    outr[idx] = accr;
    outi[idx] = acci;
}

// ============================================================
// Kernel 2: rotary on q,k (real & imag separately) then magnitude/phase.
// Layout [head][n][d]; rotary pair is (d, d^1).
// ============================================================
__global__ void rot_magphase_kernel(
    const float* __restrict__ qr, const float* __restrict__ qi,
    const float* __restrict__ kr, const float* __restrict__ ki,
    const float* __restrict__ vr, const float* __restrict__ vi,
    float* __restrict__ qm, float* __restrict__ qp,
    float* __restrict__ km, float* __restrict__ kp,
    float* __restrict__ vm, float* __restrict__ vp)
{
    const int idx = blockIdx.x * blockDim.x + threadIdx.x;
    if (idx >= HSZ) return;
    const int d = idx & (HD-1);
    const int n = (idx >> 4) % NTOK;

    // inv_freq = 10000^(-(2*(d/2))/16); cos/sin of n*inv_freq
    const float invf = __powf(10000.0f, -(float)(d & ~1) / 16.0f);
    const float fr = (float)n * invf;
    const float c = cosf(fr), s = sinf(fr);
    const int pair = idx ^ 1;
    const float sgn = (d & 1) ? 1.0f : -1.0f;  // rot[d] = (-t[d+1]) even / (t[d-1]) odd

    float a, b;
    a = qr[idx]*c + sgn*qr[pair]*s;
    b = qi[idx]*c + sgn*qi[pair]*s;
    qm[idx] = sqrtf(a*a + b*b);  qp[idx] = atan2f(b, a);

    a = kr[idx]*c + sgn*kr[pair]*s;
    b = ki[idx]*c + sgn*ki[pair]*s;
    km[idx] = sqrtf(a*a + b*b);  kp[idx] = atan2f(b, a);

    a = vr[idx];  b = vi[idx];
    vm[idx] = sqrtf(a*a + b*b);  vp[idx] = atan2f(b, a);
}

// ============================================================
// Kernel 3: qkm/qkp = scale * (Q 16xK=16) x (K^T) via f32 WMMA.
// 4 waves per block, one wave per 16-row tile. grid: (36, NH), block 128.
// A frag:  lane l, vgpr j : M=l&15, K=4c + j + 2*(l>>4)
// B frag:  lane l, vgpr j : N=l&15, K=4c + j + 2*(l>>4)   (B element = km[col=tile16+N][K])
// C frag:  lane l, vgpr r : M=r+8*(l>>4), N=l&15
//
// qkm is re-read twice later (softmax stats + prob regen): regular stores so
// lines stay L2-resident (170 MB < 192 MB L2).
// qkp is write-once: non-temporal stores so it streams to HBM without
// evicting qkm from L2.
// ============================================================
__global__ __launch_bounds__(128) void qk_wmma_kernel(
    const float* __restrict__ qm, const float* __restrict__ qp,
    const float* __restrict__ km, const float* __restrict__ kp,
    float* __restrict__ qkm, float* __restrict__ qkp)
{
    const int lane = threadIdx.x & 31;
    const int wave = threadIdx.x >> 5;
    const int tile = blockIdx.x * 4 + wave;        // 0..143
    const int head = blockIdx.y;
    const int rowl = lane & 15;
    const int half = lane >> 4;
    const int row  = tile * 16 + rowl;
    const float scale = 0.144337567297406f;        // 1/sqrt(48)

    const float* qmh = qm + (size_t)head * NTOK * HD;
    const float* qph = qp + (size_t)head * NTOK * HD;
    const float* kmh = km + (size_t)head * NTOK * HD;
    const float* kph = kp + (size_t)head * NTOK * HD;

    v2f a_m[4], a_p[4];
    #pragma unroll
    for (int c = 0; c < 4; ++c) {
        const int k0 = 4*c + 2*half;
        a_m[c] = *(const v2f*)&qmh[row*HD + k0];
        a_p[c] = *(const v2f*)&qph[row*HD + k0];
    }

    for (int j = 0; j < NTOK/16; ++j) {
        const int col = j*16 + rowl;               // this lane's B column
        v8f accm = {0.f,0.f,0.f,0.f,0.f,0.f,0.f,0.f};
        v8f accp = {0.f,0.f,0.f,0.f,0.f,0.f,0.f,0.f};
        #pragma unroll
        for (int c = 0; c < 4; ++c) {
            const int k0 = 4*c + 2*half;
            const v2f b_m = *(const v2f*)&kmh[col*HD + k0];
            const v2f b_p = *(const v2f*)&kph[col*HD + k0];
            accm = wmma_k4(a_m[c], b_m, accm);
            accp = wmma_k4(a_p[c], b_p, accp);
        }
        #pragma unroll
        for (int r = 0; r < 8; ++r) {
            const int m = r + 8*half;
            const size_t off = ((size_t)(head*NTOK + tile*16 + m)) * NTOK + j*16 + rowl;
            qkm[off] = accm[r] * scale;                                  // RT: keep in L2
            __builtin_nontemporal_store(accp[r] * scale, &qkp[off]);     // NT: stream out
        }
    }
}

// ============================================================
// Kernel 4: per-row softmax stats over |qkm|: rowmax and 1/sum(exp).
// One wave per row. grid: NH*NTOK, block 32.  (qkm reads should hit L2.)
// ============================================================
__global__ __launch_bounds__(32) void softmax_stats_kernel(
    const float* __restrict__ qkm, float* __restrict__ stats)
{
    const int row  = blockIdx.x;                    // head*NTOK + n
    const int lane = threadIdx.x;
    const float* r = qkm + (size_t)row * NTOK;

    float m = -1e30f;
    for (int c = lane; c < NTOK; c += 32) m = fmaxf(m, fabsf(r[c]));
    #pragma unroll
    for (int off = 16; off > 0; off >>= 1) m = fmaxf(m, __shfl_xor(m, off, 32));

    float s = 0.0f;
    for (int c = lane; c < NTOK; c += 32) s += __expf(fabsf(r[c]) - m);
    #pragma unroll
    for (int off = 16; off > 0; off >>= 1) s += __shfl_xor(s, off, 32);

    if (lane == 0) { stats[row*2] = m; stats[row*2+1] = 1.0f / s; }
}

// ============================================================
// Kernel 5: out = softmax(|qkm|) @ {vm,vp}, then a_r/a_i = mag*cos/sin(phase),
// written merged to NCHW [CHN][NTOK]. K = 2304 via 576 chained K=4 WMMAs.
// Probabilities are computed elementwise directly into A-fragment lanes.
// grid: (36, NH), block 128 (4 waves, one tile each).
// ============================================================
__global__ __launch_bounds__(128) void av_wmma_kernel(
    const float* __restrict__ qkm, const float* __restrict__ stats,
    const float* __restrict__ vm, const float* __restrict__ vp,
    float* __restrict__ ar, float* __restrict__ ai)
{
    const int lane = threadIdx.x & 31;
    const int wave = threadIdx.x >> 5;
    const int tile = blockIdx.x * 4 + wave;
    const int head = blockIdx.y;
    const int rowl = lane & 15;
    const int half = lane >> 4;
    const int row  = tile * 16 + rowl;

    const float* srow = qkm + ((size_t)(head*NTOK + row)) * NTOK;
    const float rmax = stats[(head*NTOK + row)*2];
    const float rinv = stats[(head*NTOK + row)*2 + 1];
    const float* vmh = vm + (size_t)head * NTOK * HD;
    const float* vph = vp + (size_t)head * NTOK * HD;

    v8f accm = {0.f,0.f,0.f,0.f,0.f,0.f,0.f,0.f};
    v8f accp = {0.f,0.f,0.f,0.f,0.f,0.f,0.f,0.f};

    for (int j = 0; j < NTOK; j += 4) {
        const int k0 = j + 2*half;                 // K coords for this lane (k0, k0+1)
        const v2f sv = *(const v2f*)&srow[k0];
        v2f a;
        a.x = __expf(fabsf(sv.x) - rmax) * rinv;
        a.y = __expf(fabsf(sv.y) - rmax) * rinv;
        v2f bm, bp;                                // B: N=rowl (=d), K rows k0,k0+1
        bm.x = vmh[(size_t)k0    *HD + rowl];
        bm.y = vmh[(size_t)(k0+1)*HD + rowl];
        bp.x = vph[(size_t)k0    *HD + rowl];
        bp.y = vph[(size_t)(k0+1)*HD + rowl];
        accm = wmma_k4(a, bm, accm);
        accp = wmma_k4(a, bp, accp);
    }

    #pragma unroll
    for (int r = 0; r < 8; ++r) {
        const int n = tile*16 + r + 8*half;        // token index (M)
        const int d = rowl;                        // head-dim (N)
        const float mag = accm[r], ph = accp[r];
        const int cidx = head*16 + d;              // merged channel
        ar[(size_t)cidx * NTOK + n] = mag * cosf(ph);
        ai[(size_t)cidx * NTOK + n] = mag * sinf(ph);
    }
}

// ============================================================
// Kernel 6: output complex conv, NCHW [CHN][NTOK] in/out.
// o_r/o_i are write-once final outputs -> non-temporal stores.
// grid: (NTOK/256, CHN), block 256
// ============================================================
__global__ void conv_o_kernel(
    const float* __restrict__ ar, const float* __restrict__ ai,
    const float* __restrict__ wr, const float* __restrict__ wi,
    const float* __restrict__ br, const float* __restrict__ bi,
    float* __restrict__ o_r, float* __restrict__ o_i)
{
    const int o = blockIdx.y;
    const int p = blockIdx.x * blockDim.x + threadIdx.x;
    const int y = p / IW, x = p % IW;

    float accr = br[o];
    float acci = bi[o];
    for (int i = 0; i < CHN; ++i) {
        const float* xrp = ar + i * (IH*IW);
        const float* xip = ai + i * (IH*IW);
        const float* wrp = wr + (o * CHN + i) * 9;
        const float* wip = wi + (o * CHN + i) * 9;
        #pragma unroll
        for (int kh = 0; kh < 3; ++kh) {
            const int yy = y + kh - 1;
            if (yy < 0 || yy >= IH) continue;
            #pragma unroll
            for (int kw = 0; kw < 3; ++kw) {
                const int xx = x + kw - 1;
                if (xx < 0 || xx >= IW) continue;
                const float xrv = xrp[yy*IW + xx];
                const float xiv = xip[yy*IW + xx];
                const float wrv = wrp[kh*3 + kw];
                const float wiv = wip[kh*3 + kw];
                accr = fmaf(xrv, wrv, accr);
                accr = fmaf(-xiv, wiv, accr);
                acci = fmaf(xrv, wiv, acci);
                acci = fmaf(xiv, wrv, acci);
            }
        }
    }
    __builtin_nontemporal_store(accr, &o_r[(size_t)o * NTOK + p]);
    __builtin_nontemporal_store(acci, &o_i[(size_t)o * NTOK + p]);
}

// ============================================================
extern "C" void kernel_launch(void* const* d_in, const int* in_sizes, int n_in,
                              void* d_out, int out_size, void* d_ws, size_t ws_size,
                              hipStream_t stream)
{
    (void)in_sizes; (void)n_in; (void)out_size; (void)ws_size;
    // setup_inputs order: x_real,x_imag, then (w_r,w_i,b_r,b_i) for q,k,v,o
    const float* xr  = (const float*)d_in[0];
    const float* xi  = (const float*)d_in[1];
    const float* wqr = (const float*)d_in[2];  const float* wqi = (const float*)d_in[3];
    const float* bqr = (const float*)d_in[4];  const float* bqi = (const float*)d_in[5];
    const float* wkr = (const float*)d_in[6];  const float* wki = (const float*)d_in[7];
    const float* bkr = (const float*)d_in[8];  const float* bki = (const float*)d_in[9];
    const float* wvr = (const float*)d_in[10]; const float* wvi = (const float*)d_in[11];
    const float* bvr = (const float*)d_in[12]; const float* bvi = (const float*)d_in[13];
    const float* wor = (const float*)d_in[14]; const float* woi = (const float*)d_in[15];
    const float* bor = (const float*)d_in[16]; const float* boi = (const float*)d_in[17];

    // d_out layout: o_r[294912], o_i[294912], qkm[8*2304^2], qkp[8*2304^2]
    float* out  = (float*)d_out;
    float* o_r  = out;
    float* o_i  = out + HSZ;
    float* qkm  = out + 2*(size_t)HSZ;
    float* qkp  = qkm + (size_t)NH * NTOK * NTOK;

    // workspace carve (floats): 14 planes of HSZ + stats
    float* ws = (float*)d_ws;
    float* qr = ws + 0*(size_t)HSZ;  float* qi = ws + 1*(size_t)HSZ;
    float* kr = ws + 2*(size_t)HSZ;  float* ki = ws + 3*(size_t)HSZ;
    float* vr = ws + 4*(size_t)HSZ;  float* vi = ws + 5*(size_t)HSZ;
    float* qm = ws + 6*(size_t)HSZ;  float* qp = ws + 7*(size_t)HSZ;
    float* km = ws + 8*(size_t)HSZ;  float* kp = ws + 9*(size_t)HSZ;
    float* vm = ws + 10*(size_t)HSZ; float* vp = ws + 11*(size_t)HSZ;
    float* ar = ws + 12*(size_t)HSZ; float* ai = ws + 13*(size_t)HSZ;
    float* stats = ws + 14*(size_t)HSZ;          // [NH*NTOK][2]

    conv_qkv_kernel<<<dim3(NTOK/256, CHN, 3), 256, 0, stream>>>(
        xr, xi, wqr, wqi, bqr, bqi, wkr, wki, bkr, bki, wvr, wvi, bvr, bvi,
        qr, qi, kr, ki, vr, vi);

    rot_magphase_kernel<<<(HSZ + 255)/256, 256, 0, stream>>>(
        qr, qi, kr, ki, vr, vi, qm, qp, km, kp, vm, vp);

    qk_wmma_kernel<<<dim3((NTOK/16)/4, NH), 128, 0, stream>>>(
        qm, qp, km, kp, qkm, qkp);

    softmax_stats_kernel<<<NH*NTOK, 32, 0, stream>>>(qkm, stats);

    av_wmma_kernel<<<dim3((NTOK/16)/4, NH), 128, 0, stream>>>(
        qkm, stats, vm, vp, ar, ai);

    conv_o_kernel<<<dim3(NTOK/256, CHN), 256, 0, stream>>>(
        ar, ai, wor, woi, bor, boi, o_r, o_i);
}